// MemN2N_82686710382739
// MI455X (gfx1250) — compile-verified
//
#include <hip/hip_runtime.h>
#include <hip/hip_bf16.h>

// Problem constants (from reference)
#define BB   256
#define VV   50000
#define EE   512
#define SS   3
#define MM   200
#define HOPS 3
#define CC   7
#define DD   1536            // S*E
#define WPAD 16              // fc_w padded to 16 rows for WMMA

// enc[d] = (d+1)/768  (derived from position_encoding(1, D))
__device__ __forceinline__ float enc_of(int d) {
    return (float)(d + 1) * (1.0f / 768.0f);
}

// ---------------------------------------------------------------------------
// Kernel 0: zero-pad fc_w (7 x D) into wpad (16 x D) so the WMMA B-fragment
// loads are unconditional (no divergent predicated loads in the hot loop).
// ---------------------------------------------------------------------------
__global__ __launch_bounds__(256)
void pad_fcw_kernel(const float* __restrict__ fc_w, float* __restrict__ wpad)
{
    const int i = blockIdx.x * 256 + threadIdx.x;   // 16*1536 = 24576 elems
    if (i < WPAD * DD) {
        const int r = i / DD;
        wpad[i] = (r < CC) ? fc_w[i] : 0.0f;
    }
}

// ---------------------------------------------------------------------------
// Kernel 1: fused MemN2N attention hops.  One block per batch element.
// 256 threads = 8 wave32s.  u kept in LDS across hops; emb re-gathered from
// L2 (emb = 102MB fits in the 192MB L2, so re-gather beats materializing the
// 315MB "m" tensor).  Vectorized b128/b64 gathers on the dominant streams.
// ---------------------------------------------------------------------------
__global__ __launch_bounds__(256)
void memn2n_attn_kernel(const int*   __restrict__ stories,  // B*M*S
                        const int*   __restrict__ queries,  // B*S
                        const float* __restrict__ emb,      // V*E
                        float*       __restrict__ u_out)    // B*D
{
    __shared__ __align__(16) float u_s[DD];    // running controller state u
    __shared__ __align__(16) float ue_s[DD];   // u * enc (refreshed per hop)
    __shared__ int   idx_s[MM * SS];
    __shared__ float prob_s[MM];

    const int b    = blockIdx.x;
    const int tid  = threadIdx.x;
    const int lane = tid & 31;
    const int wave = tid >> 5;      // 0..7

    // --- load story word indices for this batch (hop-invariant) ---
    for (int i = tid; i < MM * SS; i += 256)
        idx_s[i] = stories[b * (MM * SS) + i];

    // --- u0 = emb[queries].reshape(D) * enc ---
    for (int d = tid; d < DD; d += 256) {
        const int s = d >> 9;          // d / 512
        const int e = d & 511;         // d % 512
        const float v = emb[(size_t)queries[b * SS + s] * EE + e];
        u_s[d] = v * enc_of(d);
    }
    __syncthreads();

    for (int hop = 0; hop < HOPS; ++hop) {
        // ue = u * enc   (so that  m . u  ==  emb_gather . ue)
        for (int d = tid; d < DD; d += 256)
            ue_s[d] = u_s[d] * enc_of(d);
        __syncthreads();

        // --- dotted[mm] = sum_d m[b,mm,d] * u[b,d] ; one memory per wave ---
        // float4 loads: lane covers 16B chunks at 512B stride (coalesced).
        for (int mm = wave; mm < MM; mm += 8) {
            float acc = 0.0f;
            #pragma unroll
            for (int s = 0; s < SS; ++s) {
                const float* __restrict__ row = emb + (size_t)idx_s[mm * SS + s] * EE;
                const float* __restrict__ ue  = ue_s + s * EE;
                #pragma unroll
                for (int e4 = lane * 4; e4 < EE; e4 += 128) {
                    const float4 rv = *(const float4*)(row + e4);
                    const float4 uv = *(const float4*)(ue  + e4);
                    acc = fmaf(rv.x, uv.x, acc);
                    acc = fmaf(rv.y, uv.y, acc);
                    acc = fmaf(rv.z, uv.z, acc);
                    acc = fmaf(rv.w, uv.w, acc);
                }
            }
            #pragma unroll
            for (int off = 16; off > 0; off >>= 1)
                acc += __shfl_down(acc, off, 32);
            if (lane == 0) prob_s[mm] = acc;
        }
        __syncthreads();

        // --- softmax over M (wave 0 only) ---
        if (wave == 0) {
            float mx = -3.402823466e38f;
            for (int mm = lane; mm < MM; mm += 32)
                mx = fmaxf(mx, prob_s[mm]);
            #pragma unroll
            for (int off = 16; off > 0; off >>= 1)
                mx = fmaxf(mx, __shfl_xor(mx, off, 32));

            float sum = 0.0f;
            for (int mm = lane; mm < MM; mm += 32) {
                const float ex = expf(prob_s[mm] - mx);
                prob_s[mm] = ex;
                sum += ex;
            }
            #pragma unroll
            for (int off = 16; off > 0; off >>= 1)
                sum += __shfl_xor(sum, off, 32);
            const float inv = 1.0f / sum;
            for (int mm = lane; mm < MM; mm += 32)
                prob_s[mm] *= inv;
        }
        __syncthreads();

        // --- o_k[d] = sum_mm probs[mm] * m[b,mm,d];  u += o_k ---
        // Thread owns 2 consecutive dims per sentence slot (float2 loads,
        // 32 lanes x 8B = 256B contiguous per row: coalesced).
        {
            float2 a0 = {0.f, 0.f}, a1 = {0.f, 0.f}, a2 = {0.f, 0.f};
            const int e0 = tid * 2;                  // 0..510
            for (int mm = 0; mm < MM; ++mm) {
                const float p = prob_s[mm];
                const int* __restrict__ ip = &idx_s[mm * SS];
                const float2 v0 = *(const float2*)(emb + (size_t)ip[0] * EE + e0);
                const float2 v1 = *(const float2*)(emb + (size_t)ip[1] * EE + e0);
                const float2 v2 = *(const float2*)(emb + (size_t)ip[2] * EE + e0);
                a0.x = fmaf(p, v0.x, a0.x);  a0.y = fmaf(p, v0.y, a0.y);
                a1.x = fmaf(p, v1.x, a1.x);  a1.y = fmaf(p, v1.y, a1.y);
                a2.x = fmaf(p, v2.x, a2.x);  a2.y = fmaf(p, v2.y, a2.y);
            }
            const int d0 = e0;                       // s = 0
            const int d1 = EE  + e0;                 // s = 1
            const int d2 = 2 * EE + e0;              // s = 2
            u_s[d0    ] += a0.x * enc_of(d0);
            u_s[d0 + 1] += a0.y * enc_of(d0 + 1);
            u_s[d1    ] += a1.x * enc_of(d1);
            u_s[d1 + 1] += a1.y * enc_of(d1 + 1);
            u_s[d2    ] += a2.x * enc_of(d2);
            u_s[d2 + 1] += a2.y * enc_of(d2 + 1);
        }
        __syncthreads();
    }

    // --- write u to workspace for the WMMA FC kernel ---
    for (int d = tid; d < DD; d += 256)
        u_out[(size_t)b * DD + d] = u_s[d];
}

// ---------------------------------------------------------------------------
// Kernel 2: out = u @ fc_w^T + fc_b  via V_WMMA_F32_16X16X4_F32.
// One wave32 per 16-row tile of u.  K swept 4 at a time (384 WMMAs/tile).
// B matrix comes from the zero-padded wpad: no guards in the hot loop,
// one unconditional global_load_b64 per fragment.
// Fragment layouts per CDNA5 ISA 7.12.2:
//   A 16x4 : lane l -> row l%16 ; VGPR0/1 -> K = (l<16 ? 0,1 : 2,3)
//   B 4x16 : lane l -> col l%16 ; VGPR0/1 -> K = (l<16 ? 0,1 : 2,3)
//   C/D    : VGPR r -> row r + (l<16 ? 0 : 8) ; col = l%16
// ---------------------------------------------------------------------------
typedef __attribute__((ext_vector_type(2))) float v2f;
typedef __attribute__((ext_vector_type(8))) float v8f;

__global__ __launch_bounds__(32)
void memn2n_fc_wmma_kernel(const float* __restrict__ u,     // B*D
                           const float* __restrict__ wpad,  // 16*D (zero-padded)
                           const float* __restrict__ fc_b,  // C
                           float*       __restrict__ out)   // B*C
{
    const int lane  = threadIdx.x & 31;
    const int b0    = blockIdx.x * 16;
    const int row   = lane & 15;          // A row index / B col index
    const int khalf = (lane >> 4) << 1;   // 0 for lanes 0-15, 2 for 16-31

    const float* __restrict__ arow = u    + (size_t)(b0 + row) * DD + khalf;
    const float* __restrict__ brow = wpad + (size_t)row * DD + khalf;

    v8f acc = {};
    #pragma unroll 4
    for (int k = 0; k < DD; k += 4) {
        const float2 av = *(const float2*)(arow + k);
        const float2 bv = *(const float2*)(brow + k);
        v2f a, bmat;
        a.x = av.x;    a.y = av.y;
        bmat.x = bv.x; bmat.y = bv.y;
        // 8 args: (neg_a, A, neg_b, B, c_mod, C, reuse_a, reuse_b)
        acc = __builtin_amdgcn_wmma_f32_16x16x4_f32(
            /*neg_a=*/false, a, /*neg_b=*/false, bmat,
            /*c_mod=*/(short)0, acc, /*reuse_a=*/false, /*reuse_b=*/false);
    }

    const int rbase = (lane >> 4) << 3;   // 0 or 8
    const int ocol  = lane & 15;
    if (ocol < CC) {
        const float bias = fc_b[ocol];
        #pragma unroll
        for (int r = 0; r < 8; ++r) {
            const int orow = rbase + r;
            out[(size_t)(b0 + orow) * CC + ocol] = acc[r] + bias;
        }
    }
}

// ---------------------------------------------------------------------------
extern "C" void kernel_launch(void* const* d_in, const int* in_sizes, int n_in,
                              void* d_out, int out_size, void* d_ws, size_t ws_size,
                              hipStream_t stream)
{
    const int*   stories = (const int*)  d_in[0];   // (B, M, S) int32
    const int*   queries = (const int*)  d_in[1];   // (B, S)    int32
    const float* emb     = (const float*)d_in[2];   // (V, E)    f32
    const float* fc_w    = (const float*)d_in[3];   // (C, D)    f32
    const float* fc_b    = (const float*)d_in[4];   // (C,)      f32
    float*       out     = (float*)d_out;           // (B, C)    f32

    // Workspace layout: [ u : B*D floats | wpad : 16*D floats ]
    float* u_ws = (float*)d_ws;
    float* wpad = u_ws + (size_t)BB * DD;

    pad_fcw_kernel<<<(WPAD * DD + 255) / 256, 256, 0, stream>>>(fc_w, wpad);
    memn2n_attn_kernel<<<BB, 256, 0, stream>>>(stories, queries, emb, u_ws);
    memn2n_fc_wmma_kernel<<<BB / 16, 32, 0, stream>>>(u_ws, wpad, fc_b, out);
}